// SimpleNet_41575283425666
// MI455X (gfx1250) — compile-verified
//
#include <hip/hip_runtime.h>

typedef float v2f __attribute__((ext_vector_type(2)));
typedef float v8f __attribute__((ext_vector_type(8)));

// ---------------------------------------------------------------------------
// Zero a float region (grid-stride).
// ---------------------------------------------------------------------------
__global__ __launch_bounds__(256) void k_zero(float* __restrict__ p, long n) {
  long i = (long)blockIdx.x * blockDim.x + threadIdx.x;
  long stride = (long)gridDim.x * blockDim.x;
  for (; i < n; i += stride) p[i] = 0.0f;
}

// ---------------------------------------------------------------------------
// Edge scatter: one wave (32 lanes) per edge.
//   aggI[dst] += featS[src]   (featS has width W)
//   aggU[src] += featD[dst]
// Optionally accumulate degree counts (layer 1 only; layer 2 reuses them).
// ---------------------------------------------------------------------------
__global__ __launch_bounds__(256) void k_scatter(
    const float* __restrict__ featS, const float* __restrict__ featD,
    const int* __restrict__ esrc, const int* __restrict__ edst,
    float* __restrict__ aggI, float* __restrict__ aggU,
    float* __restrict__ cntI, float* __restrict__ cntU,
    int E, int W, int doCount) {
  int wave = (int)((blockIdx.x * blockDim.x + threadIdx.x) >> 5);
  int lane = threadIdx.x & 31;
  if (wave >= E) return;
  int s = esrc[wave];
  int d = edst[wave];
  const float4* xs = (const float4*)(featS + (long)s * W);
  const float4* xd = (const float4*)(featD + (long)d * W);
  float* ai = aggI + (long)d * W;
  float* au = aggU + (long)s * W;
  for (int base = lane * 4; base < W; base += 128) {
    float4 a = xs[base >> 2];
    atomicAdd(ai + base + 0, a.x);
    atomicAdd(ai + base + 1, a.y);
    atomicAdd(ai + base + 2, a.z);
    atomicAdd(ai + base + 3, a.w);
    float4 b = xd[base >> 2];
    atomicAdd(au + base + 0, b.x);
    atomicAdd(au + base + 1, b.y);
    atomicAdd(au + base + 2, b.z);
    atomicAdd(au + base + 3, b.w);
  }
  if (doCount && lane == 0) {
    atomicAdd(cntI + d, 1.0f);
    atomicAdd(cntU + s, 1.0f);
  }
}

// ---------------------------------------------------------------------------
// Mean-normalize: agg[row][col] /= max(cnt[row], 1)
// ---------------------------------------------------------------------------
__global__ __launch_bounds__(256) void k_norm(float* __restrict__ agg,
                                              const float* __restrict__ cnt,
                                              long M, int W) {
  long n = M * (long)W;
  long i = (long)blockIdx.x * blockDim.x + threadIdx.x;
  long stride = (long)gridDim.x * blockDim.x;
  for (; i < n; i += stride) {
    long row = i / W;
    float c = cnt[row];
    agg[i] = agg[i] / fmaxf(c, 1.0f);
  }
}

// ---------------------------------------------------------------------------
// Fused dual-GEMM with epilogue:
//   OUT[M,N] = act(A1[M,K] @ W1[K,N] + A2[M,K] @ W2[K,N] + bias[N])
// 256-thread block = 8 waves arranged 4(M) x 2(N): block tile = 64 x 32.
// Each wave accumulates a 16x16 f32 tile via v_wmma_f32_16x16x4_f32.
// A fragment (16x4): lane l holds A[m0 + l%16][k + 2*(l/16) + {0,1}]  (float2)
// B fragment (4x16): lane l holds W[k + 2*(l/16) + {0,1}][n0 + l%16]
// C/D (16x16):       VGPR v, lanes 0-15 -> row m0+v, lanes 16-31 -> row m0+8+v
//
// Branch-free inner loop: OOB rows are CLAMPED (load a valid row, discard at
// store time) rather than predicated, so there is no EXEC manipulation and
// the B addresses strength-reduce to pointer increments.
// ---------------------------------------------------------------------------
__global__ __launch_bounds__(256) void k_gemm2(
    const float* __restrict__ A1, const float* __restrict__ W1,
    const float* __restrict__ A2, const float* __restrict__ W2,
    const float* __restrict__ bias, float* __restrict__ OUT,
    int M, int K, int N, int doRelu) {
  const int lane = threadIdx.x & 31;
  const int wave = threadIdx.x >> 5;           // 0..7
  const int wm = wave & 3;                     // 0..3 (M direction)
  const int wn = wave >> 2;                    // 0..1 (N direction)
  const int m0 = blockIdx.y * 64 + wm * 16;
  const int n0 = blockIdx.x * 32 + wn * 16;

  const int rowA  = m0 + (lane & 15);          // A row this lane feeds
  const int colB  = n0 + (lane & 15);          // B/C column this lane feeds
  const int khalf = (lane >> 4) * 2;           // 0 or 2: K sub-offset
  const int rowLd = (rowA < M) ? rowA : (M - 1);   // clamp, never predicate

  v8f acc = {};
  const long strideB = (long)4 * N;            // B row advance per k-step

  {
    const float* ap  = A1 + (long)rowLd * K + khalf;
    const float* bp0 = W1 + (long)khalf * N + colB;
    const float* bp1 = bp0 + N;
#pragma unroll 8
    for (int k = 0; k < K; k += 4) {
      v2f a = *(const v2f*)ap;
      v2f b;
      b.x = *bp0;
      b.y = *bp1;
      acc = __builtin_amdgcn_wmma_f32_16x16x4_f32(
          false, a, false, b, (short)0, acc, false, false);
      ap  += 4;
      bp0 += strideB;
      bp1 += strideB;
    }
  }
  {
    const float* ap  = A2 + (long)rowLd * K + khalf;
    const float* bp0 = W2 + (long)khalf * N + colB;
    const float* bp1 = bp0 + N;
#pragma unroll 8
    for (int k = 0; k < K; k += 4) {
      v2f a = *(const v2f*)ap;
      v2f b;
      b.x = *bp0;
      b.y = *bp1;
      acc = __builtin_amdgcn_wmma_f32_16x16x4_f32(
          false, a, false, b, (short)0, acc, false, false);
      ap  += 4;
      bp0 += strideB;
      bp1 += strideB;
    }
  }

  // Epilogue: bias (+ optional ReLU), guarded store.
  const float bv = bias[colB];
  const int rbase = m0 + (lane >> 4) * 8;
#pragma unroll
  for (int v = 0; v < 8; ++v) {
    int r = rbase + v;
    if (r < M) {
      float o = acc[v] + bv;
      if (doRelu) o = fmaxf(o, 0.0f);
      OUT[(long)r * N + colB] = o;
    }
  }
}

// ---------------------------------------------------------------------------
// Decode: out[l] = dot(zU[lbl_src[l]], zI[lbl_dst[l]]) over O=128.
// One wave per label; float4 per lane; shfl_xor tree reduce (wave32).
// ---------------------------------------------------------------------------
__global__ __launch_bounds__(256) void k_decode(
    const float* __restrict__ zU, const float* __restrict__ zI,
    const int* __restrict__ ls, const int* __restrict__ ld,
    float* __restrict__ out, int L, int O) {
  int wave = (int)((blockIdx.x * blockDim.x + threadIdx.x) >> 5);
  int lane = threadIdx.x & 31;
  if (wave >= L) return;
  const float4* a = (const float4*)(zU + (long)ls[wave] * O);
  const float4* b = (const float4*)(zI + (long)ld[wave] * O);
  float s = 0.0f;
  for (int i = lane; i * 4 < O; i += 32) {
    float4 av = a[i];
    float4 bv = b[i];
    s += av.x * bv.x + av.y * bv.y + av.z * bv.z + av.w * bv.w;
  }
#pragma unroll
  for (int off = 16; off > 0; off >>= 1) s += __shfl_xor(s, off, 32);
  if (lane == 0) out[wave] = s;
}

// ---------------------------------------------------------------------------
// Host launcher
// ---------------------------------------------------------------------------
extern "C" void kernel_launch(void* const* d_in, const int* in_sizes, int n_in,
                              void* d_out, int out_size, void* d_ws, size_t ws_size,
                              hipStream_t stream) {
  const float* x_user = (const float*)d_in[0];
  const float* x_item = (const float*)d_in[1];
  const int* edge_src = (const int*)d_in[2];
  const int* edge_dst = (const int*)d_in[3];
  const int* lbl_src  = (const int*)d_in[4];
  const int* lbl_dst  = (const int*)d_in[5];
  const float* wl1_ui = (const float*)d_in[6];
  const float* wr1_ui = (const float*)d_in[7];
  const float* b1_ui  = (const float*)d_in[8];
  const float* wl1_iu = (const float*)d_in[9];
  const float* wr1_iu = (const float*)d_in[10];
  const float* b1_iu  = (const float*)d_in[11];
  const float* wl2_ui = (const float*)d_in[12];
  const float* wr2_ui = (const float*)d_in[13];
  const float* b2_ui  = (const float*)d_in[14];
  const float* wl2_iu = (const float*)d_in[15];
  const float* wr2_iu = (const float*)d_in[16];
  const float* b2_iu  = (const float*)d_in[17];
  float* out = (float*)d_out;

  const int D = 128;
  const int NU = in_sizes[0] / D;
  const int NI = in_sizes[1] / D;
  const int E  = in_sizes[2];
  const int L  = in_sizes[4];
  const int H  = in_sizes[8];    // 256
  const int O  = in_sizes[14];   // 128

  // Workspace layout (floats), zero-region first.
  float* ws = (float*)d_ws;
  size_t off = 0;
  float* cntI  = ws + off; off += (size_t)NI;
  float* cntU  = ws + off; off += (size_t)NU;
  float* aggI  = ws + off; off += (size_t)NI * D;
  float* aggU  = ws + off; off += (size_t)NU * D;
  float* aggI2 = ws + off; off += (size_t)NI * H;
  float* aggU2 = ws + off; off += (size_t)NU * H;
  long zeroN = (long)off;               // everything above must start at 0
  float* hI = ws + off; off += (size_t)NI * H;
  float* hU = ws + off; off += (size_t)NU * H;
  float* zI = ws + off; off += (size_t)NI * O;
  float* zU = ws + off; off += (size_t)NU * O;
  (void)ws_size; (void)n_in; (void)out_size;

  // 1. Zero counts + aggregation accumulators.
  k_zero<<<4096, 256, 0, stream>>>(ws, zeroN);

  // 2. Layer-1 edge scatter (+degree counts). One wave per edge, 8 per block.
  int scatterBlocks = (E + 7) / 8;
  k_scatter<<<scatterBlocks, 256, 0, stream>>>(
      x_user, x_item, edge_src, edge_dst, aggI, aggU, cntI, cntU, E, D, 1);

  // 3. Mean normalize.
  k_norm<<<4096, 256, 0, stream>>>(aggI, cntI, NI, D);
  k_norm<<<4096, 256, 0, stream>>>(aggU, cntU, NU, D);

  // 4. Layer-1 fused GEMMs: h = relu(agg @ Wl + x @ Wr + b)   [K=D, N=H]
  dim3 g1i(H / 32, (NI + 63) / 64);
  dim3 g1u(H / 32, (NU + 63) / 64);
  k_gemm2<<<g1i, 256, 0, stream>>>(aggI, wl1_ui, x_item, wr1_ui, b1_ui, hI,
                                   NI, D, H, 1);
  k_gemm2<<<g1u, 256, 0, stream>>>(aggU, wl1_iu, x_user, wr1_iu, b1_iu, hU,
                                   NU, D, H, 1);

  // 5. Layer-2 edge scatter (width H, reuse counts).
  k_scatter<<<scatterBlocks, 256, 0, stream>>>(
      hU, hI, edge_src, edge_dst, aggI2, aggU2, cntI, cntU, E, H, 0);

  // 6. Mean normalize.
  k_norm<<<4096, 256, 0, stream>>>(aggI2, cntI, NI, H);
  k_norm<<<4096, 256, 0, stream>>>(aggU2, cntU, NU, H);

  // 7. Layer-2 fused GEMMs (no ReLU): z = agg2 @ Wl2 + h @ Wr2 + b2  [K=H, N=O]
  dim3 g2i(O / 32, (NI + 63) / 64);
  dim3 g2u(O / 32, (NU + 63) / 64);
  k_gemm2<<<g2i, 256, 0, stream>>>(aggI2, wl2_ui, hI, wr2_ui, b2_ui, zI,
                                   NI, H, O, 0);
  k_gemm2<<<g2u, 256, 0, stream>>>(aggU2, wl2_iu, hU, wr2_iu, b2_iu, zU,
                                   NU, H, O, 0);

  // 8. Decode: one wave per label.
  int decodeBlocks = (L + 7) / 8;
  k_decode<<<decodeBlocks, 256, 0, stream>>>(zU, zI, lbl_src, lbl_dst, out, L, O);
}